// PFASModule_53171695125034
// MI455X (gfx1250) — compile-verified
//
#include <hip/hip_runtime.h>
#include <hip/hip_bf16.h>
#include <math.h>
#include <stdint.h>

typedef __attribute__((ext_vector_type(2))) float v2f;
typedef __attribute__((ext_vector_type(8))) float v8f;

#define KNN 16
#define MAXB 8
#define SEG 4                 // candidate-scan segments (waves) per query tile
#define WAVES_PER_BLOCK 8     // = 2 tiles * SEG segments
#define LL (KNN + 2)          // list length incl. sentinel slots

// ---------------------------------------------------------------------------
// Kernel 0: pack coords as (x,y,z,|x|^2) float4 and build batch start offsets.
// ---------------------------------------------------------------------------
__global__ void prep_kernel(const float* __restrict__ coord,
                            const int* __restrict__ batch,
                            float4* __restrict__ pack,
                            int* __restrict__ bstart, int N) {
    int i = blockIdx.x * blockDim.x + threadIdx.x;
    if (i >= N) return;
    float x = coord[i * 3 + 0], y = coord[i * 3 + 1], z = coord[i * 3 + 2];
    pack[i] = make_float4(x, y, z, x * x + y * y + z * z);
    int b  = batch[i];
    int bp = (i == 0) ? -1 : batch[i - 1];
    for (int t = bp + 1; t <= b && t <= MAXB; ++t) bstart[t] = i;
    if (i == N - 1) {
        for (int t = b + 1; t <= MAXB; ++t) bstart[t] = N;
    }
}

// ---------------------------------------------------------------------------
// Kernel 1: kNN via WMMA gram tiles, software-pipelined candidate stream.
//   wave = (tile, segment); lanes L and L+16 each track 8 candidates/tile via
//   a branchless shift-insert network; 8 sorted lists per query merged as
//   packed u64 (d2bits<<32 | idx) tournament-min from LDS.
// ---------------------------------------------------------------------------
__global__ __launch_bounds__(256) void knn_kernel(
    const float4* __restrict__ pack, const int* __restrict__ batch,
    const int* __restrict__ bstart, float* __restrict__ lin_out,
    float* __restrict__ den_out, int N) {
    __shared__ unsigned long long sl[WAVES_PER_BLOCK][32][LL];

    const int lane = threadIdx.x & 31;
    const int wave = threadIdx.x >> 5;
    const int seg  = wave & (SEG - 1);
    int tile = blockIdx.x * (WAVES_PER_BLOCK / SEG) + (wave >> 2);
    const bool tvalid = (tile * 16 + 15) < N;
    if (!tvalid) tile = 0;              // keep EXEC uniform; suppress store later
    const int i0   = tile * 16;
    const int q    = lane & 15;
    const int half = lane >> 4;         // 0: K=0,1 / rows 0-7 ; 1: K=2,3 / rows 8-15
    const int i    = i0 + q;            // this lane's query point

    float4 qf = pack[i];
    int   qb = batch[i];
    int   lo = bstart[qb], hi = bstart[qb + 1];
    float x2q = qf.w;

    // B operand: column n = query, B[k][n] = (-2x,-2y,-2z,1)
    v2f vb;
    if (half == 0) { vb.x = -2.0f * qf.x; vb.y = -2.0f * qf.y; }
    else           { vb.x = -2.0f * qf.z; vb.y = 1.0f; }

    // register-resident sorted top-16 (squared distances, ascending)
    float dl[KNN]; int il[KNN];
#pragma unroll
    for (int t = 0; t < KNN; ++t) { dl[t] = 3.0e38f; il[t] = i; }

    // wave-uniform scan range from sorted batches, split into SEG segments
    const int bfirst = batch[i0];
    const int blast  = batch[i0 + 15];
    const int jstart = bstart[bfirst] & ~15;
    const int jend   = bstart[blast + 1];
    const int ntile  = (jend - jstart + 15) >> 4;
    const int per    = (ntile + SEG - 1) >> 2;
    const int t0     = seg * per;
    const int nt     = min(ntile, t0 + per) - t0;

    // per-lane base: lane half 0 reads (x,y), half 1 reads (z,|x|^2)
    const char* cp = (const char*)pack + (half << 3)
                   + ((size_t)(jstart + (t0 << 4) + q) << 4);
    float2 cur = *(const float2*)cp;          // safe over-read stays inside d_ws
    int j0 = jstart + (t0 << 4);

    for (int tt = 0; tt < nt; ++tt, j0 += 16) {
        cp += 256;                            // next 16-candidate tile
        float2 nxt = *(const float2*)cp;      // pipelined: in flight during top-K
        __builtin_prefetch((const void*)(cp + 256), 0, 3);
        v2f va; va.x = cur.x; va.y = cur.y;
        v8f acc = {};
        acc = __builtin_amdgcn_wmma_f32_16x16x4_f32(
            false, va, false, vb, (short)0, acc, false, false);
#pragma unroll
        for (int r = 0; r < 8; ++r) {
            int   j  = j0 + r + half * 8;
            float d2 = fmaxf(acc[r] + x2q, 0.0f);   // |xj|^2 - 2 xi.xj + |xi|^2
            bool valid = (j >= lo) & (j < hi) & (j != i);
            d2 = valid ? d2 : 3.4e38f;
            if (d2 < dl[KNN - 1]) {
                // branchless shift-insert: p = first t with d2 < dl[t]
                bool c[KNN];
#pragma unroll
                for (int t = 0; t < KNN; ++t) c[t] = (d2 < dl[t]);
#pragma unroll
                for (int t = KNN - 1; t >= 1; --t) {
                    dl[t] = c[t - 1] ? dl[t - 1] : (c[t] ? d2 : dl[t]);
                    il[t] = c[t - 1] ? il[t - 1] : (c[t] ? j  : il[t]);
                }
                if (c[0]) { dl[0] = d2; il[0] = j; }
            }
        }
        cur = nxt;
    }

    // publish sorted list as packed u64 keys (+ sentinels)
#pragma unroll
    for (int t = 0; t < KNN; ++t) {
        sl[wave][lane][t] =
            ((unsigned long long)__float_as_uint(dl[t]) << 32) |
            (unsigned long long)(unsigned)il[t];
    }
    sl[wave][lane][KNN]     = ~0ull;
    sl[wave][lane][KNN + 1] = ~0ull;
    __syncthreads();

    if (seg == 0 && half == 0) {
        const int wb = wave;   // seg==0 -> wave is group base
        int h0 = 0, h1 = 0, h2 = 0, h3 = 0, h4 = 0, h5 = 0, h6 = 0, h7 = 0;
        float md[KNN]; int mi[KNN];
#pragma unroll
        for (int t = 0; t < KNN; ++t) {
            unsigned long long v0 = sl[wb + 0][lane][h0];
            unsigned long long v1 = sl[wb + 0][lane + 16][h1];
            unsigned long long v2 = sl[wb + 1][lane][h2];
            unsigned long long v3 = sl[wb + 1][lane + 16][h3];
            unsigned long long v4 = sl[wb + 2][lane][h4];
            unsigned long long v5 = sl[wb + 2][lane + 16][h5];
            unsigned long long v6 = sl[wb + 3][lane][h6];
            unsigned long long v7 = sl[wb + 3][lane + 16][h7];
            unsigned long long m01 = v0 < v1 ? v0 : v1;
            unsigned long long m23 = v2 < v3 ? v2 : v3;
            unsigned long long m45 = v4 < v5 ? v4 : v5;
            unsigned long long m67 = v6 < v7 ? v6 : v7;
            unsigned long long m03 = m01 < m23 ? m01 : m23;
            unsigned long long m47 = m45 < m67 ? m45 : m67;
            unsigned long long mv  = m03 < m47 ? m03 : m47;
            h0 += (v0 == mv); h1 += (v1 == mv); h2 += (v2 == mv); h3 += (v3 == mv);
            h4 += (v4 == mv); h5 += (v5 == mv); h6 += (v6 == mv); h7 += (v7 == mv);
            md[t] = __uint_as_float((unsigned)(mv >> 32));
            mi[t] = (int)(unsigned)(mv & 0xFFFFFFFFull);
        }
        // neighborhood moments
        float sx = 0, sy = 0, sz = 0, sxx = 0, syy = 0, szz = 0;
        float sxy = 0, sxz = 0, syz = 0, sdist = 0;
#pragma unroll
        for (int t = 0; t < KNN; ++t) {
            float4 cf = pack[mi[t]];
            sx += cf.x; sy += cf.y; sz += cf.z;
            sxx += cf.x * cf.x; syy += cf.y * cf.y; szz += cf.z * cf.z;
            sxy += cf.x * cf.y; sxz += cf.x * cf.z; syz += cf.y * cf.z;
            sdist += sqrtf(fmaxf(md[t], 1e-12f));
        }
        const float invK = 1.0f / KNN, invKm1 = 1.0f / (KNN - 1);
        float mx = sx * invK, my = sy * invK, mz = sz * invK;
        float cxx = (sxx - KNN * mx * mx) * invKm1;
        float cyy = (syy - KNN * my * my) * invKm1;
        float czz = (szz - KNN * mz * mz) * invKm1;
        float cxy = (sxy - KNN * mx * my) * invKm1;
        float cxz = (sxz - KNN * mx * mz) * invKm1;
        float cyz = (syz - KNN * my * mz) * invKm1;
        // closed-form eigenvalues of symmetric 3x3 (descending)
        float qm = (cxx + cyy + czz) * (1.0f / 3.0f);
        float p1 = cxy * cxy + cxz * cxz + cyz * cyz;
        float a0 = cxx - qm, a1 = cyy - qm, a2 = czz - qm;
        float p2 = a0 * a0 + a1 * a1 + a2 * a2 + 2.0f * p1;
        float e1, e2, e3;
        if (p2 < 1e-18f) { e1 = e2 = e3 = qm; }
        else {
            float p  = sqrtf(p2 * (1.0f / 6.0f));
            float ip = 1.0f / p;
            float b00 = a0 * ip, b11 = a1 * ip, b22 = a2 * ip;
            float b01 = cxy * ip, b02 = cxz * ip, b12 = cyz * ip;
            float detB = b00 * (b11 * b22 - b12 * b12)
                       - b01 * (b01 * b22 - b12 * b02)
                       + b02 * (b01 * b12 - b11 * b02);
            float r = fminf(1.0f, fmaxf(-1.0f, 0.5f * detB));
            float phi = acosf(r) * (1.0f / 3.0f);
            e1 = qm + 2.0f * p * cosf(phi);
            e3 = qm + 2.0f * p * cosf(phi + 2.0943951023931953f);
            e2 = 3.0f * qm - e1 - e3;
        }
        float ssum = e1 + e2 + e3 + 1e-6f;
        float lin  = (e1 - e2 - e3) / ssum;
        float den  = 1.0f / (sdist * invK + 1e-6f);
        if (tvalid) { lin_out[i] = lin; den_out[i] = den; }
    }
}

// ---------------------------------------------------------------------------
// Kernel 2: h = feat @ w1 + b1 via WMMA f32 16x16x4 (two accumulators).
// ---------------------------------------------------------------------------
__global__ __launch_bounds__(256) void gemm_h_kernel(
    const float* __restrict__ feat, const float* __restrict__ w1,
    const float* __restrict__ b1, float* __restrict__ h, int N) {
    const int lane = threadIdx.x & 31;
    const int wave = threadIdx.x >> 5;
    int tid  = blockIdx.x * WAVES_PER_BLOCK + wave;
    int trow = tid >> 2;          // 64/16 = 4 column tiles
    int tcol = tid & 3;
    if (trow * 16 >= N) return;   // wave-uniform exit
    const int r = lane & 15, half = lane >> 4;
    const int i0 = trow * 16, n0 = tcol * 16;
    v8f acc0 = {}, acc1 = {};
#pragma unroll
    for (int k0 = 0; k0 < 64; k0 += 8) {
        int ka = k0 + half * 2;
        int kb = k0 + 4 + half * 2;
        float2 fa = *(const float2*)(feat + (size_t)(i0 + r) * 64 + ka);
        float2 fb = *(const float2*)(feat + (size_t)(i0 + r) * 64 + kb);
        v2f va, wa, vbv, wbv;
        va.x = fa.x; va.y = fa.y;
        wa.x = fb.x; wa.y = fb.y;
        vbv.x = w1[ka * 64 + n0 + r];
        vbv.y = w1[(ka + 1) * 64 + n0 + r];
        wbv.x = w1[kb * 64 + n0 + r];
        wbv.y = w1[(kb + 1) * 64 + n0 + r];
        acc0 = __builtin_amdgcn_wmma_f32_16x16x4_f32(
            false, va, false, vbv, (short)0, acc0, false, false);
        acc1 = __builtin_amdgcn_wmma_f32_16x16x4_f32(
            false, wa, false, wbv, (short)0, acc1, false, false);
    }
    float bias = b1[n0 + r];
#pragma unroll
    for (int rr = 0; rr < 8; ++rr) {
        int m = rr + half * 8;
        h[(size_t)(i0 + m) * 64 + n0 + r] = acc0[rr] + acc1[rr] + bias;
    }
}

// ---------------------------------------------------------------------------
// Kernel 3: train-mode BN statistics (one block per channel).
// ---------------------------------------------------------------------------
__global__ void bn_stats_kernel(const float* __restrict__ h,
                                float* __restrict__ mu,
                                float* __restrict__ var, int N) {
    int c = blockIdx.x;
    float s = 0.0f, s2 = 0.0f;
    for (int i = threadIdx.x; i < N; i += blockDim.x) {
        float v = h[(size_t)i * 64 + c];
        s += v; s2 += v * v;
    }
    __shared__ float rs[256], rs2[256];
    rs[threadIdx.x] = s; rs2[threadIdx.x] = s2;
    __syncthreads();
    for (int off = 128; off > 0; off >>= 1) {
        if (threadIdx.x < (unsigned)off) {
            rs[threadIdx.x]  += rs[threadIdx.x + off];
            rs2[threadIdx.x] += rs2[threadIdx.x + off];
        }
        __syncthreads();
    }
    if (threadIdx.x == 0) {
        float m = rs[0] / N;
        mu[c]  = m;
        var[c] = rs2[0] / N - m * m;   // biased, as reference
    }
}

// ---------------------------------------------------------------------------
// Kernel 4: BN apply + ReLU + 64x3 head + softmax + geometric mixing.
// ---------------------------------------------------------------------------
__global__ void finalize_kernel(
    const float* __restrict__ h, const float* __restrict__ mu,
    const float* __restrict__ var, const float* __restrict__ gamma,
    const float* __restrict__ beta, const float* __restrict__ w2,
    const float* __restrict__ b2, const float* __restrict__ lin_in,
    const float* __restrict__ den_in, float* __restrict__ out, int N) {
    int i = blockIdx.x * blockDim.x + threadIdx.x;
    if (i >= N) return;
    float l0 = b2[0], l1 = b2[1], l2 = b2[2];
#pragma unroll 8
    for (int c = 0; c < 64; ++c) {
        float v  = h[(size_t)i * 64 + c];
        float hn = (v - mu[c]) * rsqrtf(var[c] + 1e-5f) * gamma[c] + beta[c];
        hn = fmaxf(hn, 0.0f);
        l0 += hn * w2[c * 3 + 0];
        l1 += hn * w2[c * 3 + 1];
        l2 += hn * w2[c * 3 + 2];
    }
    float mxv = fmaxf(l0, fmaxf(l1, l2));
    float e0 = __expf(l0 - mxv), e1 = __expf(l1 - mxv), e2 = __expf(l2 - mxv);
    float inv = 1.0f / (e0 + e1 + e2);
    float p0 = e0 * inv, p1 = e1 * inv, p2 = e2 * inv;
    float lin = lin_in[i], den = den_in[i];
    float tower = (den * 2.0f + p0) * (1.0f / 3.0f);
    float backg = (fmaxf(1.0f - lin, 1.0f - den) + p1) * (1.0f / 3.0f);
    float line  = (lin * 2.0f + p2) * (1.0f / 3.0f);
    out[i * 3 + 0] = tower * 0.2f + backg * 0.8f + line * 0.4f + 1e-6f;
    out[i * 3 + 1] = tower * 0.2f + backg * 0.8f + line * 0.4f + 1e-6f;
    out[i * 3 + 2] = tower * 0.2f + backg * 0.8f + line * 2.0f + 1e-6f;
}

// ---------------------------------------------------------------------------
extern "C" void kernel_launch(void* const* d_in, const int* in_sizes, int n_in,
                              void* d_out, int out_size, void* d_ws, size_t ws_size,
                              hipStream_t stream) {
    const float* feat  = (const float*)d_in[0];
    const float* coord = (const float*)d_in[1];
    const int*   batch = (const int*)d_in[2];
    const float* w1    = (const float*)d_in[3];
    const float* b1    = (const float*)d_in[4];
    const float* gamma = (const float*)d_in[5];
    const float* beta  = (const float*)d_in[6];
    const float* w2    = (const float*)d_in[7];
    const float* b2    = (const float*)d_in[8];
    float* out = (float*)d_out;
    const int N = in_sizes[1] / 3;

    // workspace layout
    char* p = (char*)d_ws;
    float4* pack = (float4*)p;  p += (size_t)N * sizeof(float4);
    int* bstart  = (int*)p;     p += (MAXB + 1) * sizeof(int);
    p = (char*)(((uintptr_t)p + 255) & ~(uintptr_t)255);
    float* lin = (float*)p;     p += (size_t)N * sizeof(float);
    float* den = (float*)p;     p += (size_t)N * sizeof(float);
    float* h   = (float*)p;     p += (size_t)N * 64 * sizeof(float);
    float* mu  = (float*)p;     p += 64 * sizeof(float);
    float* var = (float*)p;     p += 64 * sizeof(float);

    prep_kernel<<<(N + 255) / 256, 256, 0, stream>>>(coord, batch, pack, bstart, N);

    int ntiles = (N + 15) / 16;
    int tiles_per_block = WAVES_PER_BLOCK / SEG;
    knn_kernel<<<(ntiles + tiles_per_block - 1) / tiles_per_block, 256, 0, stream>>>(
        pack, batch, bstart, lin, den, N);

    int htiles = ntiles * 4;
    gemm_h_kernel<<<(htiles + WAVES_PER_BLOCK - 1) / WAVES_PER_BLOCK, 256, 0, stream>>>(
        feat, w1, b1, h, N);

    bn_stats_kernel<<<64, 256, 0, stream>>>(h, mu, var, N);

    finalize_kernel<<<(N + 255) / 256, 256, 0, stream>>>(
        h, mu, var, gamma, beta, w2, b2, lin, den, out, N);
}